// Quantizer_4243427688625
// MI455X (gfx1250) — compile-verified
//
#include <hip/hip_runtime.h>
#include <hip/hip_bf16.h>
#include <stdint.h>

typedef __attribute__((ext_vector_type(2))) float v2f;
typedef __attribute__((ext_vector_type(8))) float v8f;

#define C_DIM   64
#define K_CODES 2048
#define HWSZ    1024        // H*W = 32*32
#define N_ROWS  32768       // B*H*W
#define PITCH   68          // LDS row pitch in floats (4-bank stride, 8B/16B aligned)
#define NWAVES  8
#define QUANT_ELEMS 2097152 // 32*64*32*32
#define NKT     (K_CODES / 16)

// ---- prep: codebook squared norms + zero loss accumulator ----
__global__ __launch_bounds__(256)
void vq_prep_kernel(const float* __restrict__ cb, float* __restrict__ ws) {
    int k = blockIdx.x * blockDim.x + threadIdx.x;
    if (k == 0) ws[K_CODES] = 0.0f;
    if (k < K_CODES) {
        float s = 0.0f;
        #pragma unroll 8
        for (int c = 0; c < C_DIM; ++c) {
            float v = cb[k * C_DIM + c];
            s += v * v;
        }
        ws[k] = s;
    }
}

// ---- main: WMMA distances, argmin, gather, losses ----
__global__ __launch_bounds__(256)
void vq_main_kernel(const float* __restrict__ x, const float* __restrict__ cb,
                    const float* __restrict__ enorm, float* __restrict__ loss_acc,
                    float* __restrict__ quant_out, float* __restrict__ idx_out) {
    __shared__ float xfLDS[NWAVES][16 * PITCH];   // per-wave 16x64 x tile
    __shared__ float cbLDS[2][16 * PITCH];        // double-buffered codebook tile
    __shared__ float xnLDS[NWAVES][16];           // per-row ||x||^2
    __shared__ int   bkLDS[NWAVES][16];           // per-row best code

    const int tid  = threadIdx.x;
    const int wave = tid >> 5;
    const int lane = tid & 31;
    const int r    = lane & 15;       // row (A) / code-col (B) within tile
    const int half = lane >> 4;       // 0: K-sub {0,1}, 1: K-sub {2,3}
    const int coff = half * 2;

    const int tile_base = blockIdx.x * (NWAVES * 16) + wave * 16;
    const int b   = tile_base / HWSZ;            // tiles never straddle a batch
    const int hw0 = tile_base % HWSZ;

    // ---- stage A tile from NCHW (channels-last into LDS), coalesced in hw ----
    for (int cg = 0; cg < 32; ++cg) {
        int c = cg * 2 + half;
        xfLDS[wave][r * PITCH + c] = x[((b * C_DIM + c) * HWSZ) + hw0 + r];
    }
    __syncthreads();

    // ---- per-row squared norms ----
    if (lane < 16) {
        float s = 0.0f;
        #pragma unroll
        for (int c = 0; c < C_DIM; ++c) {
            float v = xfLDS[wave][lane * PITCH + c];
            s += v * v;
        }
        xnLDS[wave][lane] = s;
    }

    // ---- preload all 16 A fragments (K=4 slices over C=64) into registers ----
    v2f areg[16];
    #pragma unroll
    for (int j = 0; j < 16; ++j)
        areg[j] = *(const v2f*)&xfLDS[wave][r * PITCH + 4 * j + coff];

    __syncthreads();
    float xn[8];
    #pragma unroll
    for (int v = 0; v < 8; ++v) xn[v] = xnLDS[wave][half * 8 + v];

    float bestd[8];
    int   bestk[8];
    #pragma unroll
    for (int v = 0; v < 8; ++v) { bestd[v] = 3.4e38f; bestk[v] = 0; }

    // ---- async-copy staging coordinates (16 B per lane per tile) ----
    const int sn = tid >> 4;            // code row 0..15
    const int sc = (tid & 15) * 4;      // channel 0,4,...,60 (4 floats, same row)
    unsigned cb_base = (unsigned)(uintptr_t)(void*)&cbLDS[0][0];
    unsigned lds_dst0 = cb_base + (unsigned)((sn * PITCH + sc) * 4);
    unsigned lds_dst1 = lds_dst0 + (unsigned)(16 * PITCH * 4);
    unsigned voff = (unsigned)(tid * 16);

    // prologue: async-stage tile 0 into buffer 0
    {
        const float* gbase = cb;
        asm volatile("global_load_async_to_lds_b128 %0, %1, %2"
                     :: "v"(lds_dst0), "v"(voff), "s"(gbase) : "memory");
        asm volatile("s_wait_asynccnt 0" ::: "memory");
    }
    __syncthreads();

    // ---- sweep codebook in 16-code tiles, double-buffered async staging ----
    for (int kt = 0; kt < NKT; ++kt) {
        const int cur = kt & 1;

        // issue async copy of next tile into the other buffer (overlaps WMMA)
        if (kt + 1 < NKT) {
            const float* gbase = cb + (kt + 1) * 1024;
            unsigned dst = cur ? lds_dst0 : lds_dst1;
            asm volatile("global_load_async_to_lds_b128 %0, %1, %2"
                         :: "v"(dst), "v"(voff), "s"(gbase) : "memory");
        }

        float en = enorm[kt * 16 + r];
        v8f acc = {0.f, 0.f, 0.f, 0.f, 0.f, 0.f, 0.f, 0.f};
        const float* curbuf = &cbLDS[cur][0];
        #pragma unroll
        for (int j = 0; j < 16; ++j) {
            v2f bfrag = *(const v2f*)&curbuf[r * PITCH + 4 * j + coff];
            acc = __builtin_amdgcn_wmma_f32_16x16x4_f32(
                false, areg[j], false, bfrag, (short)0, acc, false, false);
        }

        int kidx = kt * 16 + r;   // code index owned by this lane's N column
        #pragma unroll
        for (int v = 0; v < 8; ++v) {
            float d = xn[v] + en - 2.0f * acc[v];
            if (d < bestd[v]) { bestd[v] = d; bestk[v] = kidx; }
        }

        // own async copy done, then make it visible workgroup-wide
        asm volatile("s_wait_asynccnt 0" ::: "memory");
        __syncthreads();
    }

    // ---- argmin across the 16-lane halves (N dimension), first-min tiebreak ----
    #pragma unroll
    for (int m = 1; m < 16; m <<= 1) {
        #pragma unroll
        for (int v = 0; v < 8; ++v) {
            float od = __shfl_xor(bestd[v], m, 32);
            int   ok = __shfl_xor(bestk[v], m, 32);
            if (od < bestd[v] || (od == bestd[v] && ok < bestk[v])) {
                bestd[v] = od; bestk[v] = ok;
            }
        }
    }
    if (r == 0) {
        #pragma unroll
        for (int v = 0; v < 8; ++v) bkLDS[wave][half * 8 + v] = bestk[v];
    }
    __syncthreads();

    // ---- indices output (as float), row-major (B,H,W) flat ----
    if (lane < 16) idx_out[tile_base + lane] = (float)bkLDS[wave][lane];

    // ---- gather quantized rows, write NCHW, accumulate loss ----
    int myk = bkLDS[wave][r];
    float lsum = 0.0f;
    for (int cg = 0; cg < 32; ++cg) {
        int c = cg * 2 + half;
        float q  = cb[myk * C_DIM + c];
        float xv = xfLDS[wave][r * PITCH + c];
        float d  = q - xv;
        lsum += d * d;
        quant_out[((b * C_DIM + c) * HWSZ) + hw0 + r] = q;
    }
    #pragma unroll
    for (int m = 1; m < 32; m <<= 1) lsum += __shfl_xor(lsum, m, 32);
    if (lane == 0) atomicAdd(loss_acc, lsum);
}

// ---- finalize both loss scalars ----
__global__ void vq_fin_kernel(const float* __restrict__ acc, float* __restrict__ losses) {
    float l = acc[0] * (1.0f / (float)QUANT_ELEMS);
    losses[0] = l;   // codebook_loss
    losses[1] = l;   // commitment_loss (identical forward value)
}

extern "C" void kernel_launch(void* const* d_in, const int* in_sizes, int n_in,
                              void* d_out, int out_size, void* d_ws, size_t ws_size,
                              hipStream_t stream) {
    const float* x  = (const float*)d_in[0];
    const float* cb = (const float*)d_in[1];
    float* out = (float*)d_out;
    float* ws  = (float*)d_ws;

    float* quant  = out;                    // [0, 2097152)
    float* losses = out + QUANT_ELEMS;      // 2 scalars
    float* idxout = out + QUANT_ELEMS + 2;  // 32768 indices (as float)

    vq_prep_kernel<<<K_CODES / 256, 256, 0, stream>>>(cb, ws);
    vq_main_kernel<<<N_ROWS / (NWAVES * 16), 256, 0, stream>>>(
        x, cb, ws, ws + K_CODES, quant, idxout);
    vq_fin_kernel<<<1, 1, 0, stream>>>(ws + K_CODES, losses);
}